// ToyModel_50053548868162
// MI455X (gfx1250) — compile-verified
//
#include <hip/hip_runtime.h>

// ---------------------------------------------------------------------------
// BTT factorized linear, fused two-stage f32-WMMA kernel for gfx1250 (MI455X).
//   stage1: h[t,n,k] = sum_b x[t,n*64+b] * btt_r[n,b,k]           (k = m*2+r)
//   stage2: y[t,m,a] = sum_{n,r} h[t,n,m*2+r] * btt_l[m,n*2+r,a] + bias
// Fused over n-chunks of 8: live h slab stays in LDS, y accumulates in
// registers. x slabs double-buffered into LDS via global_load_async_to_lds_b128
// (ASYNCcnt). Weights pre-packed into WMMA lane order (one coalesced b128 per
// 2 WMMAs) and software-pipelined one step ahead so WMMAs don't stall on L2.
// ---------------------------------------------------------------------------

typedef float v2f __attribute__((ext_vector_type(2)));
typedef float v4f __attribute__((ext_vector_type(4)));
typedef float v8f __attribute__((ext_vector_type(8)));

namespace {
constexpr int kTokens = 16384;
constexpr int kDin    = 4096;   // N*Bb
constexpr int kDout   = 4096;   // M*A
constexpr int kAcols  = 64;     // A
constexpr int kBrows  = 64;     // Bb

// LDS h-chunk: hbuf[q][t][col], col stride 128->130 (bank-conflict-free b64
// reads in phase B: 16-lane groups step banks by 2, halves offset by 32).
constexpr int kHCol = 130;
// LDS x-chunk: xbuf[t][col], col stride 512->516 (t-stride % 64 banks == 4,
// so the wave's four K-columns cover 64 distinct banks -> conflict free).
constexpr int kXStr = 516;

// Packed weight sizes (in float4 quads).
constexpr int kPRQuads = 64 * 8 * 8 * 32;     // [n][kk2][ct][lane]   (2MB)
constexpr int kPLQuads = 64 * 8 * 2 * 4 * 32; // [m][c][kk2][ct][lane](2MB)
}

// --------------------------- weight repack ---------------------------------
// pr[((n*8+kk2)*8+ct)*32+lane] = { btt_r[n][8kk2+2h  ][col], [8kk2+2h+1][col],
//                                  btt_r[n][8kk2+2h+4][col], [8kk2+2h+5][col] }
// pl[(((m*8+c)*2+kk2)*4+ct)*32+lane] = same pattern on btt_l rows c*16+8kk2+2h
// with h = lane>>4, col = ct*16 + (lane&15): exactly the per-lane B operands
// of two consecutive K=4 WMMAs -> one b128 per 2 WMMAs, 512B/wave coalesced.
__global__ void btt_repack_weights(const float* __restrict__ btt_l,
                                   const float* __restrict__ btt_r,
                                   v4f* __restrict__ pr,
                                   v4f* __restrict__ pl)
{
    const int i = blockIdx.x * blockDim.x + threadIdx.x;
    if (i < kPRQuads) {
        const int lane = i & 31, ct = (i >> 5) & 7, kk2 = (i >> 8) & 7, n = i >> 11;
        const int h = lane >> 4, col = ct * 16 + (lane & 15);
        const float* base = btt_r + (size_t)n * 8192;   // 64x128
        const int k0 = 8 * kk2 + 2 * h;
        v4f o = { base[(size_t)k0 * 128 + col],
                  base[(size_t)(k0 + 1) * 128 + col],
                  base[(size_t)(k0 + 4) * 128 + col],
                  base[(size_t)(k0 + 5) * 128 + col] };
        pr[i] = o;
    } else {
        const int j = i - kPRQuads;
        const int lane = j & 31, ct = (j >> 5) & 3, kk2 = (j >> 7) & 1;
        const int c = (j >> 8) & 7, m = j >> 11;
        const int h = lane >> 4, col = ct * 16 + (lane & 15);
        const float* base = btt_l + (size_t)m * 8192;   // 128x64
        const int j0 = c * 16 + 8 * kk2 + 2 * h;
        v4f o = { base[(size_t)j0 * 64 + col],
                  base[(size_t)(j0 + 1) * 64 + col],
                  base[(size_t)(j0 + 4) * 64 + col],
                  base[(size_t)(j0 + 5) * 64 + col] };
        pl[j] = o;
    }
}

// ----------------------------- main kernel ---------------------------------
template <bool PACKED>
__global__ __launch_bounds__(512)
void btt_fused_wmma(const float* __restrict__ x,
                    const float* __restrict__ btt_l,
                    const float* __restrict__ btt_r,
                    const float* __restrict__ bias,
                    const v4f* __restrict__ pr,
                    const v4f* __restrict__ pl,
                    float* __restrict__ y)
{
    __shared__ float hbuf[8 * 16 * kHCol];   // 66,560 B
    __shared__ float xbuf[2][16 * kXStr];    // 66,048 B (double buffered)

    const int lane = threadIdx.x & 31;       // wave32
    const int wave = threadIdx.x >> 5;       // 0..15
    const int half = lane >> 4;              // WMMA half-wave
    const int l16  = lane & 15;
    const int t0   = blockIdx.x * 16;        // 16-token tile

    const int nloc   = wave >> 1;            // phase A: n_local 0..7
    const int ctb    = (wave & 1) * 4;       // phase A: col-tile base
    const int m_base = wave * 4;             // phase B: owned m-blocks

    // Issue one 16-token x 512-col x slab (32KB) into xbuf[p] via async DMA.
    // 512 threads x 4 x b128: rows are 2KB contiguous -> fully coalesced.
    auto issue_x = [&](int c, int p) {
        const float* gbase = x + (size_t)t0 * kDin + c * 512;      // uniform
        unsigned lbase = (unsigned)(size_t)(const void*)&xbuf[p][0];
        #pragma unroll
        for (int rd = 0; rd < 4; ++rd) {
            const int f  = (int)threadIdx.x + rd * 512;  // float4 id 0..2047
            const int t  = f >> 7;                       // token row 0..15
            const int c4 = f & 127;                      // float4 col
            unsigned voff  = (unsigned)((t * kDin + c4 * 4) * 4);      // bytes
            unsigned laddr = lbase + (unsigned)((t * kXStr + c4 * 4) * 4);
            asm volatile("global_load_async_to_lds_b128 %0, %1, %2"
                         :: "v"(laddr), "v"(voff), "s"(gbase) : "memory");
        }
    };

    // Pipelined phase-A weight fetch: 4 x b128 for one kk2 step of n-block n.
    auto load_a_w = [&](int n, int kk2, v4f* dst) {
        const v4f* prn = pr + (size_t)n * (8 * 8 * 32);
        #pragma unroll
        for (int j = 0; j < 4; ++j)
            dst[j] = prn[(kk2 * 8 + ctb + j) * 32 + lane];
    };
    // Pipelined phase-B weight fetch: 4 x b128 for group g=(mm,kk2) of chunk c.
    auto load_b_w = [&](int c, int g, v4f* dst) {
        const int m = m_base + (g >> 1), kk2 = g & 1;
        const v4f* plm = pl + (size_t)(m * 8 + c) * (2 * 4 * 32);
        #pragma unroll
        for (int ct = 0; ct < 4; ++ct)
            dst[ct] = plm[(kk2 * 4 + ct) * 32 + lane];
    };

    // Stage-2 accumulators: 4 m-blocks x 4 col-tiles = 128 VGPRs.
    v8f acc[4][4];
    #pragma unroll
    for (int i = 0; i < 4; ++i)
        #pragma unroll
        for (int j = 0; j < 4; ++j)
            acc[i][j] = (v8f)0.0f;

    v4f bac[4], bbc[4];                      // in-flight weight staging
    issue_x(0, 0);
    if constexpr (PACKED) load_a_w(nloc, 0, bac);   // chunk 0, kk2=0

    for (int c = 0; c < 8; ++c) {
        const int p = c & 1;
        asm volatile("s_wait_asynccnt 0x0" ::: "memory");  // own slab done
        __syncthreads();                                   // everyone's slab done
        if (c < 7) issue_x(c + 1, 1 - p);                  // overlap with compute

        // ---------------- Phase A: stage 1 for n = c*8 + nloc ----------------
        const int n = c * 8 + nloc;
        const float* xl = &xbuf[p][l16 * kXStr + nloc * 64 + 2 * half];

        v8f hacc[4];
        #pragma unroll
        for (int j = 0; j < 4; ++j) hacc[j] = (v8f)0.0f;

        if constexpr (PACKED) {
            #pragma unroll
            for (int kk2 = 0; kk2 < 8; ++kk2) {
                v4f ban[4];
                if (kk2 < 7) load_a_w(n, kk2 + 1, ban);   // fetch next step
                v2f a0 = *reinterpret_cast<const v2f*>(xl + kk2 * 8);
                v2f a1 = *reinterpret_cast<const v2f*>(xl + kk2 * 8 + 4);
                #pragma unroll
                for (int j = 0; j < 4; ++j) {
                    hacc[j] = __builtin_amdgcn_wmma_f32_16x16x4_f32(
                        false, a0, false, bac[j].xy, (short)0, hacc[j], false, false);
                    hacc[j] = __builtin_amdgcn_wmma_f32_16x16x4_f32(
                        false, a1, false, bac[j].zw, (short)0, hacc[j], false, false);
                }
                if (kk2 < 7) {
                    #pragma unroll
                    for (int j = 0; j < 4; ++j) bac[j] = ban[j];
                }
            }
            load_b_w(c, 0, bbc);   // prefetch phase-B group 0 across barrier
        } else {
            const float* br = btt_r + (size_t)n * (kBrows * 128);
            for (int kk = 0; kk < 16; ++kk) {
                const int k = kk * 4 + 2 * half;
                v2f a = *reinterpret_cast<const v2f*>(xl + kk * 4);
                #pragma unroll
                for (int j = 0; j < 4; ++j) {
                    const int col = (ctb + j) * 16 + l16;
                    v2f b;
                    b.x = br[(size_t)k * 128 + col];
                    b.y = br[(size_t)(k + 1) * 128 + col];
                    hacc[j] = __builtin_amdgcn_wmma_f32_16x16x4_f32(
                        false, a, false, b, (short)0, hacc[j], false, false);
                }
            }
        }

        // Spill stage-1 tiles: hbuf[nloc][t][col]
        #pragma unroll
        for (int j = 0; j < 4; ++j) {
            const int col = (ctb + j) * 16 + l16;
            #pragma unroll
            for (int i = 0; i < 8; ++i) {
                const int t = i + 8 * half;
                hbuf[(nloc * 16 + t) * kHCol + col] = hacc[j][i];
            }
        }
        __syncthreads();

        // -------- Phase B: stage-2 partial, K-slab = 16 (8 n x rank 2) -------
        if constexpr (PACKED) {
            #pragma unroll
            for (int g = 0; g < 8; ++g) {
                const int mm = g >> 1, kk2 = g & 1;
                const int m = m_base + mm;
                v4f bbn[4];
                if (g < 7) load_b_w(c, g + 1, bbn);       // fetch next group
                const int q0 = 4 * kk2 + half;            // q for kk=2kk2
                v2f a0 = *reinterpret_cast<const v2f*>(
                    &hbuf[(q0 * 16 + l16) * kHCol + 2 * m]);
                v2f a1 = *reinterpret_cast<const v2f*>(
                    &hbuf[((q0 + 2) * 16 + l16) * kHCol + 2 * m]);
                #pragma unroll
                for (int ct = 0; ct < 4; ++ct) {
                    acc[mm][ct] = __builtin_amdgcn_wmma_f32_16x16x4_f32(
                        false, a0, false, bbc[ct].xy, (short)0, acc[mm][ct], false, false);
                    acc[mm][ct] = __builtin_amdgcn_wmma_f32_16x16x4_f32(
                        false, a1, false, bbc[ct].zw, (short)0, acc[mm][ct], false, false);
                }
                if (g < 7) {
                    #pragma unroll
                    for (int ct = 0; ct < 4; ++ct) bbc[ct] = bbn[ct];
                }
            }
            if (c < 7) load_a_w((c + 1) * 8 + nloc, 0, bac);  // next chunk's A weights
        } else {
            #pragma unroll
            for (int mm = 0; mm < 4; ++mm) {
                const int m = m_base + mm;
                const float* bl = btt_l + (size_t)m * (128 * kAcols)
                                        + (size_t)(c * 16) * kAcols;
                #pragma unroll
                for (int kk = 0; kk < 4; ++kk) {
                    const int q = 2 * kk + half;
                    v2f a2 = *reinterpret_cast<const v2f*>(
                        &hbuf[(q * 16 + l16) * kHCol + 2 * m]);
                    const int jg = kk * 4 + 2 * half;
                    #pragma unroll
                    for (int ct = 0; ct < 4; ++ct) {
                        const int col = ct * 16 + l16;
                        v2f b2;
                        b2.x = bl[(size_t)jg * kAcols + col];
                        b2.y = bl[(size_t)(jg + 1) * kAcols + col];
                        acc[mm][ct] = __builtin_amdgcn_wmma_f32_16x16x4_f32(
                            false, a2, false, b2, (short)0, acc[mm][ct], false, false);
                    }
                }
            }
        }
        __syncthreads();   // hbuf + xbuf[p] consumed before next chunk reuses
    }

    // ------------------------- Epilogue: bias + store ------------------------
    #pragma unroll
    for (int mm = 0; mm < 4; ++mm) {
        #pragma unroll
        for (int ct = 0; ct < 4; ++ct) {
            const int g = (m_base + mm) * kAcols + ct * 16 + l16;
            const float bv = bias[g];
            #pragma unroll
            for (int i = 0; i < 8; ++i) {
                const int t = t0 + i + 8 * half;
                // y is pure streaming output: non-temporal, keep L2 for weights
                __builtin_nontemporal_store(acc[mm][ct][i] + bv,
                                            &y[(size_t)t * kDout + g]);
            }
        }
    }
}

extern "C" void kernel_launch(void* const* d_in, const int* in_sizes, int n_in,
                              void* d_out, int out_size, void* d_ws, size_t ws_size,
                              hipStream_t stream) {
    (void)in_sizes; (void)n_in; (void)out_size;
    const float* x     = (const float*)d_in[0];
    const float* btt_l = (const float*)d_in[1];
    const float* btt_r = (const float*)d_in[2];
    const float* bias  = (const float*)d_in[3];
    float* y = (float*)d_out;

    const size_t packed_bytes = (size_t)(kPRQuads + kPLQuads) * sizeof(v4f); // 4MB
    dim3 grid(kTokens / 16), block(512);

    if (ws_size >= packed_bytes) {
        v4f* pr = (v4f*)d_ws;
        v4f* pl = pr + kPRQuads;
        hipLaunchKernelGGL(btt_repack_weights, dim3((kPRQuads + kPLQuads) / 256),
                           dim3(256), 0, stream, btt_l, btt_r, pr, pl);
        hipLaunchKernelGGL((btt_fused_wmma<true>), grid, block, 0, stream,
                           x, btt_l, btt_r, bias, pr, pl, y);
    } else {
        hipLaunchKernelGGL((btt_fused_wmma<false>), grid, block, 0, stream,
                           x, btt_l, btt_r, bias, (const v4f*)nullptr,
                           (const v4f*)nullptr, y);
    }
}